// StarTopologyFCN_59485297050172
// MI455X (gfx1250) — compile-verified
//
#include <hip/hip_runtime.h>
#include <hip/hip_bf16.h>

// ---------------------------------------------------------------------------
// StarTopologyFCN for MI455X (gfx1250, wave32, WMMA)
//
// Strategy: counting-sort batch rows by domain (8 domains) so each 16-row
// tile is single-domain; each layer is then a dense WMMA GEMM with the fused
// weight W_d = sw * dw[d] staged in LDS. fp32 throughout via
// V_WMMA_F32_16X16X4_F32 (problem is HBM/latency bound: ~11MB traffic,
// ~0.7 GFLOP -> ~0.5us at 23.3 TB/s; precision downcast buys nothing).
//
// gfx1250 paths: v_wmma_f32_16x16x4_f32 compute; A-tiles staged with
// GLOBAL_LOAD_ASYNC_TO_LDS_B128 (ASYNCcnt) when the builtin is available.
// ---------------------------------------------------------------------------

#define BATCH      2048
#define IN_DIM     512
#define NDOM       8
#define MAX_TILES  (BATCH / 16 + NDOM)   // 136 (padding adds <16 rows/domain)
#define PAD_B      (MAX_TILES * 16)      // 2176

typedef __attribute__((ext_vector_type(2))) float v2f;
typedef __attribute__((ext_vector_type(8))) float v8f;
typedef __attribute__((ext_vector_type(4))) int   v4i;

#if __has_builtin(__builtin_amdgcn_global_load_async_to_lds_b128)
#define HAVE_ASYNC_LDS 1
#else
#define HAVE_ASYNC_LDS 0
#endif

__device__ __forceinline__ void wait_asynccnt0() {
#if __has_builtin(__builtin_amdgcn_s_wait_asynccnt)
  __builtin_amdgcn_s_wait_asynccnt(0);
#elif defined(__gfx1250__)
  asm volatile("s_wait_asynccnt 0" ::: "memory");
#endif
}

// ---------------------------------------------------------------------------
// Kernel 1: counting sort by domain, padded to 16-row tiles.
//   perm[PAD_B]          : source batch row for each permuted slot, -1 = pad
//   tileDomain[MAX_TILES]: domain of each 16-row tile, -1 = unused tile
// ---------------------------------------------------------------------------
__global__ __launch_bounds__(256) void sort_kernel(
    const int* __restrict__ dom, int* __restrict__ perm,
    int* __restrict__ tileDomain) {
  __shared__ int cnt[NDOM];
  __shared__ int start[NDOM];
  __shared__ int cur[NDOM];
  const int t = threadIdx.x;

  if (t < NDOM) cnt[t] = 0;
  __syncthreads();
  for (int b = t; b < BATCH; b += 256) atomicAdd(&cnt[dom[b]], 1);
  __syncthreads();
  if (t == 0) {
    int off = 0;
    for (int d = 0; d < NDOM; ++d) {
      start[d] = off;
      cur[d] = 0;
      off += (cnt[d] + 15) & ~15;  // pad each domain segment to 16 rows
    }
  }
  __syncthreads();
  for (int i = t; i < PAD_B; i += 256) perm[i] = -1;
  for (int i = t; i < MAX_TILES; i += 256) {
    const int base = i * 16;
    int dd = -1;
    for (int d = 0; d < NDOM; ++d)
      if (cnt[d] > 0 && base >= start[d] && base < start[d] + cnt[d]) dd = d;
    tileDomain[i] = dd;
  }
  __syncthreads();
  for (int b = t; b < BATCH; b += 256) {
    const int d = dom[b];
    const int pos = start[d] + atomicAdd(&cur[d], 1);
    perm[pos] = b;  // intra-domain order is irrelevant: rows are independent
  }
}

// ---------------------------------------------------------------------------
// Kernel 2: one fused layer. Block = 128 threads = 4 wave32s.
//   blockIdx.x = 16-row batch tile, blockIdx.y * 4 + wave = 16-col out tile.
//   K processed in 128-wide chunks staged in LDS.
// ---------------------------------------------------------------------------
template <int FIN, int FOUT, bool GATHER_IN, bool SCATTER_OUT>
__global__ __launch_bounds__(128) void layer_kernel(
    const float* __restrict__ in,   // GATHER_IN: [BATCH,FIN] ; else [PAD_B,FIN]
    const float* __restrict__ sw,   // [FIN,FOUT]
    const float* __restrict__ dw,   // [NDOM,FIN,FOUT]
    const float* __restrict__ sb,   // [FOUT]
    const float* __restrict__ db,   // [NDOM,FOUT]
    const int* __restrict__ perm,
    const int* __restrict__ tileDomain,
    float* __restrict__ out) {      // SCATTER_OUT: [BATCH,FOUT] ; else [PAD_B,FOUT]
  constexpr int KC = 128;           // K-chunk (divides 512/256/128)
  constexpr int AP = KC + 4;        // A row pitch in LDS (bank-conflict pad)
  static_assert(FIN % KC == 0, "KC must divide FIN");

  __shared__ float lA[16 * AP];              // 16 x KC activations
  __shared__ float lW[4][KC * 16];           // per-wave fused weight slice
  __shared__ int rowSrc[16];

  const int tile = blockIdx.x;
  const int d = tileDomain[tile];
  if (d < 0) return;  // unused padded tile

  const int tid = threadIdx.x;
  const int lane = tid & 31;
  const int wv = tid >> 5;
  const int colBase = (blockIdx.y * 4 + wv) * 16;

  if (tid < 16) {
    const int p = perm[tile * 16 + tid];
    rowSrc[tid] = GATHER_IN ? p : (p >= 0 ? tile * 16 + tid : -1);
  }

#if HAVE_ASYNC_LDS
  // Pre-zero the A tile once: pad rows are never async-loaded and must read
  // as zero in every chunk. (Valid rows are fully overwritten per chunk.)
  for (int i = tid; i < 16 * AP; i += 128) lA[i] = 0.f;
#endif

  v8f acc = {};
  const int laneHi = lane >> 4;   // 0: lanes 0-15, 1: lanes 16-31
  const int laneLo = lane & 15;

  // Precomputed LDS operand bases -> affine kk offsets fold into DS immediates.
  const float* aptr = &lA[laneLo * AP + laneHi * 2];
  const float* bptr = &lW[wv][laneHi * 16 + laneLo];

  for (int kc0 = 0; kc0 < FIN; kc0 += KC) {
    __syncthreads();  // previous chunk consumed (also covers rowSrc/zero init)

    // ---- stage A tile: 16 rows x KC ----
    for (int it = 0; it < (16 * KC / 4) / 128; ++it) {
      const int e = tid + it * 128;          // 0 .. 511
      const int j = e >> 5;                  // row 0..15
      const int c4 = e & 31;                 // float4 index 0..31
      const int r = rowSrc[j];
#if HAVE_ASYNC_LDS
      if (r >= 0) {
        const float* src = &in[(size_t)r * FIN + kc0 + c4 * 4];
        float* dst = &lA[j * AP + c4 * 4];
        // 16B/lane direct global->LDS DMA, tracked by ASYNCcnt.
        __builtin_amdgcn_global_load_async_to_lds_b128(
            (v4i*)src, (v4i*)dst, 0, 0);
      }
#else
      float4 v = make_float4(0.f, 0.f, 0.f, 0.f);
      if (r >= 0) v = *(const float4*)&in[(size_t)r * FIN + kc0 + c4 * 4];
      *(float4*)&lA[j * AP + c4 * 4] = v;
#endif
    }

    // ---- stage fused weights: this wave's 16-col slice, KC rows ----
    // (must go through VALU: element-wise sw*dw fusion; each element is
    //  consumed exactly once by the WMMA loop, so fuse-at-stage is optimal)
    {
      const int c = (lane & 3) * 4;          // col within tile, float4
      for (int pass = 0; pass < KC / 8; ++pass) {
        const int kloc = pass * 8 + (lane >> 2);
        const size_t gk = (size_t)(kc0 + kloc);
        const float4 s = *(const float4*)&sw[gk * FOUT + colBase + c];
        const float4 w = *(const float4*)&dw[((size_t)d * FIN + gk) * FOUT + colBase + c];
        float4 f;
        f.x = s.x * w.x; f.y = s.y * w.y; f.z = s.z * w.z; f.w = s.w * w.w;
        *(float4*)&lW[wv][kloc * 16 + c] = f;
      }
    }

    // prefetch next chunk's weights into cache while we compute
    if (kc0 + KC < FIN) {
      __builtin_prefetch(&sw[(size_t)(kc0 + KC) * FOUT + colBase], 0, 0);
      __builtin_prefetch(&dw[((size_t)d * FIN + kc0 + KC) * FOUT + colBase], 0, 0);
    }

#if HAVE_ASYNC_LDS
    wait_asynccnt0();   // own async LDS writes visible before the barrier
#endif
    __syncthreads();

    // ---- WMMA over the chunk: 32 x v_wmma_f32_16x16x4_f32 ----
    // A 16x4 f32 layout: lanes 0-15 hold K=kk+0/kk+1, lanes 16-31 K=kk+2/kk+3.
    // B 4x16 layout: V0 rows K=kk+laneHi, V1 rows K=kk+2+laneHi, N = laneLo.
#pragma unroll 8
    for (int kk = 0; kk < KC; kk += 4) {
      v2f a = *(const v2f*)&aptr[kk];        // ds_load_b64
      v2f b;
      b[0] = bptr[kk * 16];
      b[1] = bptr[(kk + 2) * 16];
      acc = __builtin_amdgcn_wmma_f32_16x16x4_f32(
          false, a, false, b, (short)0, acc, false, false);
    }
  }

  // ---- epilogue: bias + ReLU + store ----
  // C/D layout: VGPR v, lane -> M = v + 8*laneHi, N = laneLo.
  const int col = colBase + laneLo;
  const float bias = sb[col] + db[d * FOUT + col];
#pragma unroll
  for (int v = 0; v < 8; ++v) {
    const int m = v + 8 * laneHi;
    float val = acc[v] + bias;
    val = val > 0.f ? val : 0.f;
    if (SCATTER_OUT) {
      const int r = perm[tile * 16 + m];
      if (r >= 0) out[(size_t)r * FOUT + col] = val;
    } else {
      out[(size_t)(tile * 16 + m) * FOUT + col] = val;
    }
  }
}

// ---------------------------------------------------------------------------
extern "C" void kernel_launch(void* const* d_in, const int* in_sizes, int n_in,
                              void* d_out, int out_size, void* d_ws, size_t ws_size,
                              hipStream_t stream) {
  const float* x   = (const float*)d_in[0];
  const int* dom   = (const int*)d_in[1];
  const float* sw0 = (const float*)d_in[2];
  const float* dw0 = (const float*)d_in[3];
  const float* sb0 = (const float*)d_in[4];
  const float* db0 = (const float*)d_in[5];
  const float* sw1 = (const float*)d_in[6];
  const float* dw1 = (const float*)d_in[7];
  const float* sb1 = (const float*)d_in[8];
  const float* db1 = (const float*)d_in[9];
  const float* sw2 = (const float*)d_in[10];
  const float* dw2 = (const float*)d_in[11];
  const float* sb2 = (const float*)d_in[12];
  const float* db2 = (const float*)d_in[13];
  float* out = (float*)d_out;

  char* ws = (char*)d_ws;
  int* perm       = (int*)(ws);                        // PAD_B ints
  int* tileDomain = (int*)(ws + PAD_B * sizeof(int));  // MAX_TILES ints
  float* h1 = (float*)(ws + 16384);                    // [PAD_B, 256]
  float* h2 = h1 + (size_t)PAD_B * 256;                // [PAD_B, 128]

  sort_kernel<<<1, 256, 0, stream>>>(dom, perm, tileDomain);

  layer_kernel<512, 256, true, false>
      <<<dim3(MAX_TILES, 4), 128, 0, stream>>>(x, sw0, dw0, sb0, db0,
                                               perm, tileDomain, h1);
  layer_kernel<256, 128, false, false>
      <<<dim3(MAX_TILES, 2), 128, 0, stream>>>(h1, sw1, dw1, sb1, db1,
                                               perm, tileDomain, h2);
  layer_kernel<128, 64, false, true>
      <<<dim3(MAX_TILES, 1), 128, 0, stream>>>(h2, sw2, dw2, sb2, db2,
                                               perm, tileDomain, out);
}